// QMDN_4028679324122
// MI455X (gfx1250) — compile-verified
//
#include <hip/hip_runtime.h>
#include <math.h>

typedef __attribute__((ext_vector_type(2))) float v2f;
typedef __attribute__((ext_vector_type(8))) float v8f;

#define LAYERS 8
#define EPS_CLIP 1e-7f
#define PI_OVER_4 0.78539816339744831f

// ---------------------------------------------------------------------------
// Setup kernel: fold the x-independent circuit (8 layers of Rot+CNOTs) into a
// 4x4 complex matrix U per weight set, then emit a real 24x4 coefficient
// matrix M into d_ws:
//   rows 8*set + 2k   : Re coefficients of amplitude k
//   rows 8*set + 2k+1 : Im coefficients of amplitude k
// acting on v = (c0c1, c0s1, s0c1, s0s1):
//   Re_k = ur[k][0]*a + ui[k][1]*b + ui[k][2]*c - ur[k][3]*d
//   Im_k = ui[k][0]*a - ur[k][1]*b - ur[k][2]*c - ui[k][3]*d
// ---------------------------------------------------------------------------
__device__ inline void rot2x2(const float* w, float Rr[2][2], float Ri[2][2]) {
    float phi = w[0], th = w[1], om = w[2];
    float ct = cosf(0.5f * th), st = sinf(0.5f * th);
    float ap = 0.5f * (phi + om), am = 0.5f * (phi - om);
    float cap = cosf(ap), sap = sinf(ap), cam = cosf(am), sam = sinf(am);
    // R00 = e^{-i ap} ct ; R01 = -e^{+i am} st ; R10 = e^{-i am} st ; R11 = e^{+i ap} ct
    Rr[0][0] =  ct * cap;  Ri[0][0] = -ct * sap;
    Rr[0][1] = -st * cam;  Ri[0][1] = -st * sam;
    Rr[1][0] =  st * cam;  Ri[1][0] = -st * sam;
    Rr[1][1] =  ct * cap;  Ri[1][1] =  ct * sap;
}

__global__ void qmdn_setup(const float* __restrict__ wp,
                           const float* __restrict__ wm,
                           const float* __restrict__ ws,
                           float* __restrict__ M) {
    int set = threadIdx.x;
    if (set >= 3) return;
    const float* w = (set == 0) ? wp : ((set == 1) ? wm : ws);

    float Ur[4][4], Ui[4][4];
    for (int i = 0; i < 4; ++i)
        for (int j = 0; j < 4; ++j) { Ur[i][j] = (i == j) ? 1.0f : 0.0f; Ui[i][j] = 0.0f; }

    for (int l = 0; l < LAYERS; ++l) {
        float R0r[2][2], R0i[2][2], R1r[2][2], R1i[2][2];
        rot2x2(w + (l * 2 + 0) * 3, R0r, R0i);
        rot2x2(w + (l * 2 + 1) * 3, R1r, R1i);
        // G = R0 (wire0) kron R1 (wire1), flat index = wire0*2 + wire1
        float Gr[4][4], Gi[4][4];
        for (int i = 0; i < 2; ++i)
            for (int j = 0; j < 2; ++j)
                for (int k = 0; k < 2; ++k)
                    for (int m = 0; m < 2; ++m) {
                        Gr[i * 2 + j][k * 2 + m] = R0r[i][k] * R1r[j][m] - R0i[i][k] * R1i[j][m];
                        Gi[i * 2 + j][k * 2 + m] = R0r[i][k] * R1i[j][m] + R0i[i][k] * R1r[j][m];
                    }
        // T = G * U (complex 4x4 matmul)
        float Tr[4][4], Ti[4][4];
        for (int i = 0; i < 4; ++i)
            for (int j = 0; j < 4; ++j) {
                float ar = 0.0f, ai = 0.0f;
                for (int k = 0; k < 4; ++k) {
                    ar += Gr[i][k] * Ur[k][j] - Gi[i][k] * Ui[k][j];
                    ai += Gr[i][k] * Ui[k][j] + Gi[i][k] * Ur[k][j];
                }
                Tr[i][j] = ar; Ti[i][j] = ai;
            }
        // CNOT(0->1): swap state components 2,3 -> swap rows 2,3
        for (int j = 0; j < 4; ++j) {
            float t;
            t = Tr[2][j]; Tr[2][j] = Tr[3][j]; Tr[3][j] = t;
            t = Ti[2][j]; Ti[2][j] = Ti[3][j]; Ti[3][j] = t;
        }
        // CNOT(1->0): swap state components 1,3 -> swap rows 1,3
        for (int j = 0; j < 4; ++j) {
            float t;
            t = Tr[1][j]; Tr[1][j] = Tr[3][j]; Tr[3][j] = t;
            t = Ti[1][j]; Ti[1][j] = Ti[3][j]; Ti[3][j] = t;
        }
        for (int i = 0; i < 4; ++i)
            for (int j = 0; j < 4; ++j) { Ur[i][j] = Tr[i][j]; Ui[i][j] = Ti[i][j]; }
    }

    int off = set * 8;
    for (int k = 0; k < 4; ++k) {
        float* re = M + (off + 2 * k) * 4;
        float* im = M + (off + 2 * k + 1) * 4;
        re[0] =  Ur[k][0]; re[1] =  Ui[k][1]; re[2] =  Ui[k][2]; re[3] = -Ur[k][3];
        im[0] =  Ui[k][0]; im[1] = -Ur[k][1]; im[2] = -Ur[k][2]; im[3] = -Ui[k][3];
    }
}

// ---------------------------------------------------------------------------
// Main kernel
// ---------------------------------------------------------------------------
__device__ inline float clip01(float p) {
    return fminf(fmaxf(p, EPS_CLIP), 1.0f - EPS_CLIP);
}

__device__ inline void qmdn_scalar(float xi, const float* __restrict__ M,
                                   float* __restrict__ out, size_t B, size_t e) {
    float s0, c0, s1, c1;
    sincosf(0.5f * xi, &s0, &c0);
    sincosf(PI_OVER_4 * xi, &s1, &c1);
    float v0 = c0 * c1, v1 = c0 * s1, v2 = s0 * c1, v3 = s0 * s1;
    float p[3][4];
    for (int s = 0; s < 3; ++s)
        for (int k = 0; k < 4; ++k) {
            const float* r0 = M + (s * 8 + 2 * k) * 4;
            const float* r1 = r0 + 4;
            float re = r0[0] * v0 + r0[1] * v1 + r0[2] * v2 + r0[3] * v3;
            float im = r1[0] * v0 + r1[1] * v1 + r1[2] * v2 + r1[3] * v3;
            p[s][k] = re * re + im * im;
        }
    float inv = 1.0f / (p[0][0] + p[0][1] + p[0][2]);
    float lm3 = logf(clip01(p[1][3]));
    float ls3 = logf(clip01(p[2][3]));
    for (int k = 0; k < 3; ++k) {
        out[e * 3 + k]               = p[0][k] * inv;
        out[3 * B + e * 3 + k]       = logf(clip01(p[1][k])) - lm3;
        out[6 * B + e * 3 + k]       = expf(logf(clip01(p[2][k])) - ls3);
    }
}

__global__ void qmdn_main(const float* __restrict__ x,
                          const float* __restrict__ M,
                          float* __restrict__ out, int B) {
    const int lane   = threadIdx.x & 31;
    const int gtid   = blockIdx.x * blockDim.x + threadIdx.x;
    const int gwave  = gtid >> 5;
    const int nwaves = (gridDim.x * blockDim.x) >> 5;
    const size_t Bs  = (size_t)B;

    // A-matrix (16x4 f32) lane layout per ISA: VGPR0 = {K=0 | K=2}, VGPR1 = {K=1 | K=3};
    // lanes 0-15 and 16-31 both index M=0..15.
    const int mrow = lane & 15;
    const int kb   = (lane < 16) ? 0 : 2;
    v2f a_lo, a_hi;
    a_lo.x = M[mrow * 4 + kb];
    a_lo.y = M[mrow * 4 + kb + 1];
    a_hi.x = (mrow < 8) ? M[(16 + mrow) * 4 + kb]     : 0.0f;
    a_hi.y = (mrow < 8) ? M[(16 + mrow) * 4 + kb + 1] : 0.0f;

    const int Bfull = B & ~31;

    for (int base = gwave * 32; base + 32 <= B; base += nwaves * 32) {
        float xi = x[base + lane];
        float s0, c0, s1, c1;
        sincosf(0.5f * xi, &s0, &c0);
        sincosf(PI_OVER_4 * xi, &s1, &c1);
        float va = c0 * c1, vb = c0 * s1, vc = s0 * c1, vd = s0 * s1;

#pragma unroll
        for (int g = 0; g < 2; ++g) {
            // Build B (4x16): column n = element (base + g*16 + n), K rows = (a,b,c,d).
            // Assumed B layout mirrors A: VGPR0 = {K=0 | K=2}, VGPR1 = {K=1 | K=3}, N = lane&15.
            int src = g * 16 + mrow;
            float ta = __shfl(va, src, 32);
            float tb = __shfl(vb, src, 32);
            float tc = __shfl(vc, src, 32);
            float td = __shfl(vd, src, 32);
            v2f bm;
            bm.x = (lane < 16) ? ta : tc;
            bm.y = (lane < 16) ? tb : td;

            v8f cz = {0.f, 0.f, 0.f, 0.f, 0.f, 0.f, 0.f, 0.f};
            // D1: rows 0-15 = sets p (0-7) and m (8-15); D2: rows 16-23 = set s.
            v8f d1 = __builtin_amdgcn_wmma_f32_16x16x4_f32(
                false, a_lo, false, bm, (short)0, cz, false, false);
            v8f d2 = __builtin_amdgcn_wmma_f32_16x16x4_f32(
                false, a_hi, false, bm, (short)0, cz, false, false);

            // C/D layout: VGPR r -> lanes 0-15: M=r, N=lane; lanes 16-31: M=r+8, N=lane-16.
            size_t e = (size_t)base + (size_t)g * 16 + (size_t)mrow;
            if (lane < 16) {
                // lane n holds rows 0..7 of D1 (set p) and rows 16..23 via D2 (set s)
                float pp0 = d1[0] * d1[0] + d1[1] * d1[1];
                float pp1 = d1[2] * d1[2] + d1[3] * d1[3];
                float pp2 = d1[4] * d1[4] + d1[5] * d1[5];
                float inv = 1.0f / (pp0 + pp1 + pp2);
                out[e * 3 + 0] = pp0 * inv;
                out[e * 3 + 1] = pp1 * inv;
                out[e * 3 + 2] = pp2 * inv;

                float ps0 = d2[0] * d2[0] + d2[1] * d2[1];
                float ps1 = d2[2] * d2[2] + d2[3] * d2[3];
                float ps2 = d2[4] * d2[4] + d2[5] * d2[5];
                float ps3 = d2[6] * d2[6] + d2[7] * d2[7];
                float ls3 = logf(clip01(ps3));
                out[6 * Bs + e * 3 + 0] = expf(logf(clip01(ps0)) - ls3);
                out[6 * Bs + e * 3 + 1] = expf(logf(clip01(ps1)) - ls3);
                out[6 * Bs + e * 3 + 2] = expf(logf(clip01(ps2)) - ls3);
            } else {
                // lane n+16 holds rows 8..15 of D1 (set m)
                float pm0 = d1[0] * d1[0] + d1[1] * d1[1];
                float pm1 = d1[2] * d1[2] + d1[3] * d1[3];
                float pm2 = d1[4] * d1[4] + d1[5] * d1[5];
                float pm3 = d1[6] * d1[6] + d1[7] * d1[7];
                float lm3 = logf(clip01(pm3));
                out[3 * Bs + e * 3 + 0] = logf(clip01(pm0)) - lm3;
                out[3 * Bs + e * 3 + 1] = logf(clip01(pm1)) - lm3;
                out[3 * Bs + e * 3 + 2] = logf(clip01(pm2)) - lm3;
            }
        }
    }

    // Scalar tail for B % 32 leftovers (empty when B = 2^20).
    for (int e = Bfull + gtid; e < B; e += gridDim.x * blockDim.x) {
        qmdn_scalar(x[e], M, out, Bs, (size_t)e);
    }
}

extern "C" void kernel_launch(void* const* d_in, const int* in_sizes, int n_in,
                              void* d_out, int out_size, void* d_ws, size_t ws_size,
                              hipStream_t stream) {
    const float* x  = (const float*)d_in[0];
    const float* wp = (const float*)d_in[1];
    const float* wm = (const float*)d_in[2];
    const float* ws = (const float*)d_in[3];
    float* out = (float*)d_out;
    float* M   = (float*)d_ws;   // 24x4 f32 coefficient matrix (384 bytes)
    int B = in_sizes[0];

    qmdn_setup<<<1, 4, 0, stream>>>(wp, wm, ws, M);

    int blocks = (B + 255) / 256;
    if (blocks < 1) blocks = 1;
    if (blocks > 65535) blocks = 65535;  // grid-stride covers the rest
    qmdn_main<<<blocks, 256, 0, stream>>>(x, M, out, B);
}